// NeuralPoints_49984829390880
// MI455X (gfx1250) — compile-verified
//
#include <hip/hip_runtime.h>
#include <hip/hip_bf16.h>

// ---- problem constants from the reference ----
#define RQ   4096
#define SRQ  32
#define KQ   8
#define NEARP 2.0f
#define FARP  6.0f

typedef __attribute__((ext_vector_type(2))) float v2f;
typedef __attribute__((ext_vector_type(8))) float v8f;

// ---------------------------------------------------------------------------
// Kernel A: point_xyz_pers for all N points via V_WMMA_F32_16X16X4_F32.
//   cam_i = sum_j (p_j - campos_j) * camrot[j][i]   (einsum 'bnj,bji->bni')
// A (16x4, M=points, K=xyz|pad): lanes 0-15 hold (K=0,K=1)=(x,y) for M=lane,
//                                lanes 16-31 hold (K=2,K=3)=(z,0) for M=lane-16.
// B (4x16): V0 lanes0-15 = row K=0, lanes16-31 = K=1; V1 = K=2 / K=3.
// D (16x16): VGPR r = row M=r (lanes 0-15) / M=r+8 (lanes 16-31), N=lane%16.
// Only columns N=0..2 are meaningful; spill the tile to LDS, then lanes 0-15
// each finish one point (perspective divide) and store pers xyz.
// ---------------------------------------------------------------------------
__global__ void np_pers_wmma_kernel(const float* __restrict__ points_xyz,
                                    const float* __restrict__ camrot,  // [9]
                                    const float* __restrict__ campos,  // [3]
                                    float* __restrict__ pers_out,      // [N*3]
                                    int N, int ntiles) {
    __shared__ float tile[8][16 * 16];
    const int wave = threadIdx.x >> 5;
    const int lane = threadIdx.x & 31;
    const int t    = blockIdx.x * 8 + wave;
    const bool active = (t < ntiles);
    const int  nn = lane & 15;          // column / point-in-tile index
    const int  m  = (t << 4) + nn;      // point index this lane loads

    float sx = 0.f, sy = 0.f, sz = 0.f;
    if (active && m < N) {
        sx = points_xyz[m * 3 + 0] - campos[0];
        sy = points_xyz[m * 3 + 1] - campos[1];
        sz = points_xyz[m * 3 + 2] - campos[2];
    }
    v2f a;
    if (lane < 16) { a.x = sx; a.y = sy; }   // K=0, K=1
    else           { a.x = sz; a.y = 0.f; }  // K=2, K=3(pad)

    // B = camrot zero-padded into 4x16 (rows K=0..2 valid, cols N=0..2 valid)
    float b0 = 0.f, b1 = 0.f;
    if (nn < 3) {
        const int k0 = (lane < 16) ? 0 : 1;   // V0 row
        const int k1 = (lane < 16) ? 2 : 3;   // V1 row
        b0 = camrot[k0 * 3 + nn];
        if (k1 < 3) b1 = camrot[k1 * 3 + nn];
    }
    v2f b; b.x = b0; b.y = b1;

    v8f c = {0.f, 0.f, 0.f, 0.f, 0.f, 0.f, 0.f, 0.f};
    // EXEC must be all-ones for WMMA: execute unconditionally (inactive waves
    // computed on zeros), predicate only loads/stores.
    v8f d = __builtin_amdgcn_wmma_f32_16x16x4_f32(
        false, a, false, b, (short)0, c, false, false);

#pragma unroll
    for (int r = 0; r < 8; ++r) {
        const int M = (lane < 16) ? r : (r + 8);
        tile[wave][M * 16 + nn] = d[r];
    }
    __syncthreads();

    if (lane < 16) {
        const int mm = (t << 4) + lane;
        if (active && mm < N) {
            const float cx = tile[wave][lane * 16 + 0];
            const float cy = tile[wave][lane * 16 + 1];
            const float cz = tile[wave][lane * 16 + 2];
            pers_out[mm * 3 + 0] = cx / cz;
            pers_out[mm * 3 + 1] = cy / cz;
            pers_out[mm * 3 + 2] = cz;
        }
    }
}

// ---------------------------------------------------------------------------
// Kernel B: the irregular gather. One thread per (r,s,k) sample-neighbor.
// L2-resident gathers (tables ~76MB < 192MB L2); embedding rows are 128B
// aligned -> float4 loads; output sub-arrays start at odd float offsets so
// stores are scalar dwords (fully coalesced across lanes anyway).
// ---------------------------------------------------------------------------
__global__ void np_gather_kernel(const float* __restrict__ points_xyz,
                                 const float* __restrict__ points_emb,
                                 const float* __restrict__ points_conf,
                                 const float* __restrict__ points_dir,
                                 const float* __restrict__ points_color,
                                 const float* __restrict__ camrot,
                                 const float* __restrict__ campos,
                                 const int*   __restrict__ sample_pidx,
                                 const float* __restrict__ pers_ws,
                                 int use_ws, int N, int C,
                                 float* __restrict__ out,
                                 long long o_color, long long o_dir,
                                 long long o_emb,   long long o_pers,
                                 long long o_world, long long o_conf,
                                 long long o_mask) {
    const long long total = (long long)RQ * SRQ * KQ;
    const long long g = (long long)blockIdx.x * blockDim.x + threadIdx.x;
    if (g >= total) return;

    const int idx   = sample_pidx[g];
    const bool mask = (idx >= 0);
    const int  ci   = mask ? idx : 0;          // jnp.clip(pidx, 0)

    out[o_mask + g] = mask ? 1.0f : 0.0f;
    out[o_conf + g] = points_conf[ci];

    const float px = points_xyz[ci * 3 + 0];
    const float py = points_xyz[ci * 3 + 1];
    const float pz = points_xyz[ci * 3 + 2];
    out[o_world + g * 3 + 0] = px;
    out[o_world + g * 3 + 1] = py;
    out[o_world + g * 3 + 2] = pz;

    float ppx, ppy, ppz;
    if (use_ws) {
        ppx = pers_ws[ci * 3 + 0];
        ppy = pers_ws[ci * 3 + 1];
        ppz = pers_ws[ci * 3 + 2];
    } else {
        const float sx = px - campos[0];
        const float sy = py - campos[1];
        const float sz = pz - campos[2];
        const float cx = sx * camrot[0] + sy * camrot[3] + sz * camrot[6];
        const float cy = sx * camrot[1] + sy * camrot[4] + sz * camrot[7];
        const float cz = sx * camrot[2] + sy * camrot[5] + sz * camrot[8];
        ppx = cx / cz; ppy = cy / cz; ppz = cz;
    }
    out[o_pers + g * 3 + 0] = ppx;
    out[o_pers + g * 3 + 1] = ppy;
    out[o_pers + g * 3 + 2] = ppz;

    out[o_color + g * 3 + 0] = points_color[ci * 3 + 0];
    out[o_color + g * 3 + 1] = points_color[ci * 3 + 1];
    out[o_color + g * 3 + 2] = points_color[ci * 3 + 2];

    out[o_dir + g * 3 + 0] = points_dir[ci * 3 + 0];
    out[o_dir + g * 3 + 1] = points_dir[ci * 3 + 1];
    out[o_dir + g * 3 + 2] = points_dir[ci * 3 + 2];

    const long long eo = o_emb + g * (long long)C;
    if (C == 32) {
        const float4* erow =
            reinterpret_cast<const float4*>(points_emb + (long long)ci * 32);
#pragma unroll
        for (int c4 = 0; c4 < 8; ++c4) {
            const float4 v = erow[c4];       // global_load_b128
            out[eo + c4 * 4 + 0] = v.x;
            out[eo + c4 * 4 + 1] = v.y;
            out[eo + c4 * 4 + 2] = v.z;
            out[eo + c4 * 4 + 3] = v.w;
        }
    } else {
        const float* erow = points_emb + (long long)ci * C;
        for (int cc = 0; cc < C; ++cc) out[eo + cc] = erow[cc];
    }
}

// ---------------------------------------------------------------------------
// Kernel C: per-(ray,sample) outputs + tiny copies (Rw2c, ray_mask).
// sample_loc = pers(t * (dir @ camrot)); t cancels in x/y of the divide.
// ---------------------------------------------------------------------------
__global__ void np_persample_kernel(const float* __restrict__ ray_dirs, // [R*3]
                                    const float* __restrict__ campos,
                                    const float* __restrict__ camrot,
                                    const float* __restrict__ Rw2c,     // [9]
                                    const int*   __restrict__ ray_mask, // [R]
                                    float* __restrict__ out,
                                    long long o_loc, long long o_locw,
                                    long long o_rdir, long long o_rw2c,
                                    long long o_rmask) {
    const int g = blockIdx.x * blockDim.x + threadIdx.x;
    if (g < 9)  out[o_rw2c + g]  = Rw2c[g];
    if (g < RQ) out[o_rmask + g] = (float)ray_mask[g];

    const int total = RQ * SRQ;
    if (g >= total) return;
    const int r = g / SRQ;
    const int s = g - r * SRQ;

    const float step = (FARP - NEARP) / (float)SRQ;
    const float t = NEARP + ((float)s + 0.5f) * step;

    const float dx = ray_dirs[r * 3 + 0];
    const float dy = ray_dirs[r * 3 + 1];
    const float dz = ray_dirs[r * 3 + 2];

    out[o_locw + g * 3 + 0] = campos[0] + dx * t;
    out[o_locw + g * 3 + 1] = campos[1] + dy * t;
    out[o_locw + g * 3 + 2] = campos[2] + dz * t;

    out[o_rdir + g * 3 + 0] = dx;
    out[o_rdir + g * 3 + 1] = dy;
    out[o_rdir + g * 3 + 2] = dz;

    const float cx = dx * camrot[0] + dy * camrot[3] + dz * camrot[6];
    const float cy = dx * camrot[1] + dy * camrot[4] + dz * camrot[7];
    const float cz = dx * camrot[2] + dy * camrot[5] + dz * camrot[8];
    out[o_loc + g * 3 + 0] = cx / cz;       // (t*cx)/(t*cz)
    out[o_loc + g * 3 + 1] = cy / cz;
    out[o_loc + g * 3 + 2] = t * cz;
}

extern "C" void kernel_launch(void* const* d_in, const int* in_sizes, int n_in,
                              void* d_out, int out_size, void* d_ws, size_t ws_size,
                              hipStream_t stream) {
    const float* points_xyz   = (const float*)d_in[0];
    const float* points_emb   = (const float*)d_in[1];
    const float* points_conf  = (const float*)d_in[2];
    const float* points_dir   = (const float*)d_in[3];
    const float* points_color = (const float*)d_in[4];
    const float* points_Rw2c  = (const float*)d_in[5];
    const float* camrot       = (const float*)d_in[6];
    const float* campos       = (const float*)d_in[7];
    const float* ray_dirs     = (const float*)d_in[8];
    const int*   sample_pidx  = (const int*)d_in[9];
    const int*   ray_mask     = (const int*)d_in[10];
    float*       out          = (float*)d_out;

    const int N = in_sizes[0] / 3;
    const int C = in_sizes[1] / N;

    // flat output offsets in reference return order
    const long long nrsk = (long long)RQ * SRQ * KQ;   // 1,048,576
    const long long nrs  = (long long)RQ * SRQ;        // 131,072
    const long long o_color = 0;
    const long long o_rw2c  = o_color + nrsk * 3;
    const long long o_dir   = o_rw2c + 9;
    const long long o_emb   = o_dir + nrsk * 3;
    const long long o_pers  = o_emb + nrsk * (long long)C;
    const long long o_world = o_pers + nrsk * 3;
    const long long o_conf  = o_world + nrsk * 3;
    const long long o_loc   = o_conf + nrsk;
    const long long o_locw  = o_loc + nrs * 3;
    const long long o_mask  = o_locw + nrs * 3;
    const long long o_rdir  = o_mask + nrsk;
    const long long o_rmask = o_rdir + nrs * 3;

    const int use_ws = (ws_size >= (size_t)N * 3 * sizeof(float)) ? 1 : 0;

    if (use_ws) {
        const int ntiles  = (N + 15) >> 4;        // 16 points per wave-tile
        const int blocksA = (ntiles + 7) >> 3;    // 8 waves per 256-thread block
        np_pers_wmma_kernel<<<blocksA, 256, 0, stream>>>(
            points_xyz, camrot, campos, (float*)d_ws, N, ntiles);
    }

    const int blocksB = (int)((nrsk + 255) / 256);
    np_gather_kernel<<<blocksB, 256, 0, stream>>>(
        points_xyz, points_emb, points_conf, points_dir, points_color,
        camrot, campos, sample_pidx, (const float*)d_ws, use_ws, N, C,
        out, o_color, o_dir, o_emb, o_pers, o_world, o_conf, o_mask);

    const int blocksC = (int)((nrs + 255) / 256);
    np_persample_kernel<<<blocksC, 256, 0, stream>>>(
        ray_dirs, campos, camrot, points_Rw2c, ray_mask, out,
        o_loc, o_locw, o_rdir, o_rw2c, o_rmask);
}